// NMS_20796231647610
// MI455X (gfx1250) — compile-verified
//
#include <hip/hip_runtime.h>
#include <cstdint>

// ---------------------------------------------------------------------------
// YOLO post-process + greedy NMS for MI455X (gfx1250, wave32).
//   grid  = B blocks (one block per batch image)
//   block = 1024 threads (32 waves), each thread owns ITEMS=11 boxes in VGPRs
// Phase 1: stream class logits (NT loads), argmax80 per box; meanwhile the
//          per-batch objectness vector p is DMA'd into LDS by the TDM
//          (tensor_load_to_lds) and by async global->LDS loads.
// Phase 2: 100 greedy iterations, all state register/LDS resident:
//          ds_bpermute butterfly argmax -> LDS cross-wave reduce ->
//          broadcast winner -> IoU suppression in registers.
// ---------------------------------------------------------------------------

#define TPB    1024
#define ITEMS  11                      // ceil(10647/1024)
#define N13    507
#define N26    2028
#define N52    8112
#define NBOX   10647
#define MAXB   100
#define NEGV   (-1e9f)

typedef __attribute__((ext_vector_type(4))) float        f32x4;
typedef __attribute__((ext_vector_type(4))) unsigned int u32x4;
typedef __attribute__((ext_vector_type(8))) unsigned int u32x8;

__device__ __forceinline__ uint32_t lds_byte_addr(const void* p) {
  // generic pointer to LDS: low 32 bits are the byte offset in the group's LDS
  return (uint32_t)(uintptr_t)p;
}

// ---- Tensor Data Mover: 1D row of n f32 elements, global -> LDS -------------
__device__ __forceinline__ void tdm_load_1d_f32(const float* gsrc,
                                                uint32_t lds_byte, uint32_t n) {
  uint64_t ga = (uint64_t)(uintptr_t)gsrc;
  u32x4 g0; u32x8 g1;
  g0[0] = 1u;                                          // count=1, user mode
  g0[1] = lds_byte;                                    // lds_addr
  g0[2] = (uint32_t)ga;                                // global_addr[31:0]
  g0[3] = (uint32_t)((ga >> 32) & 0x01FFFFFFull)       // global_addr[56:32]
        | (2u << 30);                                  // type=2 ("image")
  g1[0] = (2u << 16);                                  // data_size=4B, mask=0
  g1[1] = (n & 0xFFFFu) << 16;                         // tensor_dim0 lo16
  g1[2] = ((n >> 16) & 0xFFFFu) | (1u << 16);          // dim0 hi16, tensor_dim1=1
  g1[3] = (n & 0xFFFFu) << 16;                         // tile_dim0 = n (<65536)
  g1[4] = 1u;                                          // tile_dim1=1, tile_dim2=0
  g1[5] = n;                                           // tensor_dim0_stride lo32
  g1[6] = 0u;
  g1[7] = 0u;
  asm volatile("tensor_load_to_lds %0, %1" :: "s"(g0), "s"(g1) : "memory");
}

// ---- async global -> LDS, one dword per lane --------------------------------
__device__ __forceinline__ void async_ld_b32(uint32_t lds_byte, const float* g) {
  uint64_t ga = (uint64_t)(uintptr_t)g;
  asm volatile("global_load_async_to_lds_b32 %0, %1, off"
               :: "v"(lds_byte), "v"(ga) : "memory");
}

__device__ __forceinline__ void amax2(float& bv, int& bi, float ov, int oi) {
  if (ov > bv || (ov == bv && oi < bi)) { bv = ov; bi = oi; }  // first-max ties
}

__global__ __launch_bounds__(TPB) void yolo_nms_kernel(
    const float* __restrict__ bbox13, const float* __restrict__ p13,
    const float* __restrict__ c13,
    const float* __restrict__ bbox26, const float* __restrict__ p26,
    const float* __restrict__ c26,
    const float* __restrict__ bbox52, const float* __restrict__ p52,
    const float* __restrict__ c52,
    float* __restrict__ out) {
  __shared__ float p_lds[NBOX];     // 42,588 B objectness, DMA-staged
  __shared__ float red_v[32];
  __shared__ int   red_i[32];
  __shared__ float win[8];          // 0:score 1:idx 2..5:raw box 6:area

  const int b    = blockIdx.x;
  const int tid  = threadIdx.x;
  const int lane = tid & 31;
  const int wid  = tid >> 5;
  const uint32_t p_base = lds_byte_addr(&p_lds[0]);

  // ---- kick off TDM DMA of this batch's p slices (wave 0 only) -------------
  if (tid == 0) {
    tdm_load_1d_f32(p13 + (size_t)b * N13, p_base + 0,                 N13);
    tdm_load_1d_f32(p26 + (size_t)b * N26, p_base + (uint32_t)N13 * 4, N26);
    tdm_load_1d_f32(p52 + (size_t)b * N52, p_base + (uint32_t)(N13 + N26) * 4, N52);
  }

  float sc[ITEMS];                      // phase1: class argmax idx; then score
  float by0[ITEMS], bx0[ITEMS], by1[ITEMS], bx1[ITEMS], ar[ITEMS];

  // ---- Phase 1: boxes + argmax over 80 classes (NT streaming) --------------
#pragma unroll
  for (int k = 0; k < ITEMS; ++k) {
    const int g = tid + k * TPB;
    sc[k] = NEGV; by0[k] = bx0[k] = by1[k] = bx1[k] = 0.f; ar[k] = 0.f;
    if (g < NBOX) {
      const float *cb, *pb, *bb; int off;
      if (g < N13)            { off = b * N13 + g;               cb = c13; pb = p13; bb = bbox13; }
      else if (g < N13 + N26) { off = b * N26 + (g - N13);       cb = c26; pb = p26; bb = bbox26; }
      else                    { off = b * N52 + (g - N13 - N26); cb = c52; pb = p52; bb = bbox52; }

      async_ld_b32(p_base + (uint32_t)g * 4u, pb + (size_t)off);   // 2nd async path

      const f32x4 bx = __builtin_nontemporal_load((const f32x4*)(bb + (size_t)off * 4));
      by0[k] = bx[0]; bx0[k] = bx[1]; by1[k] = bx[2]; bx1[k] = bx[3];

      const f32x4* cp = (const f32x4*)(cb + (size_t)off * 80);
      __builtin_prefetch(cp, 0, 1);                                // global_prefetch
      f32x4 v = __builtin_nontemporal_load(cp);
      float m = v[0]; int mi = 0;
      if (v[1] > m) { m = v[1]; mi = 1; }
      if (v[2] > m) { m = v[2]; mi = 2; }
      if (v[3] > m) { m = v[3]; mi = 3; }
#pragma unroll
      for (int j = 1; j < 20; ++j) {
        f32x4 u = __builtin_nontemporal_load(cp + j);
        const int base = j * 4;
        if (u[0] > m) { m = u[0]; mi = base + 0; }
        if (u[1] > m) { m = u[1]; mi = base + 1; }
        if (u[2] > m) { m = u[2]; mi = base + 2; }
        if (u[3] > m) { m = u[3]; mi = base + 3; }
      }
      sc[k] = (float)mi;                 // stash argmax index; score in phase 2
    }
  }

  // ---- wait for both DMA paths, then score + area --------------------------
  asm volatile("s_wait_asynccnt 0x0" ::: "memory");
  if (tid == 0) __builtin_amdgcn_s_wait_tensorcnt(0);
  __syncthreads();

#pragma unroll
  for (int k = 0; k < ITEMS; ++k) {
    const int g = tid + k * TPB;
    if (g < NBOX) {
      const float s = p_lds[g] * sc[k];
      sc[k] = (s > 0.5f) ? s : NEGV;     // strict SCORE_THR
      const float y1 = fminf(by0[k], by1[k]), y2 = fmaxf(by0[k], by1[k]);
      const float x1 = fminf(bx0[k], bx1[k]), x2 = fmaxf(bx0[k], bx1[k]);
      ar[k] = (y2 - y1) * (x2 - x1);
    }
  }

  // ---- Phase 2: greedy NMS, 100 serial rounds ------------------------------
  int count = 0;
  for (int it = 0; it < MAXB; ++it) {
    // local argmax over register-resident scores
    float bv = NEGV; int bi = 0x7FFFFFFF;
#pragma unroll
    for (int k = 0; k < ITEMS; ++k) {
      const int g = tid + k * TPB;
      if (g < NBOX) amax2(bv, bi, sc[k], g);
    }
    // wave32 butterfly via ds_bpermute
#pragma unroll
    for (int off = 16; off > 0; off >>= 1) {
      const int sl = (lane ^ off) << 2;
      const float ov = __int_as_float(__builtin_amdgcn_ds_bpermute(sl, __float_as_int(bv)));
      const int   oi = __builtin_amdgcn_ds_bpermute(sl, bi);
      amax2(bv, bi, ov, oi);
    }
    if (lane == 0) { red_v[wid] = bv; red_i[wid] = bi; }
    __syncthreads();
    if (wid == 0) {
      float v2 = red_v[lane]; int i2 = red_i[lane];
#pragma unroll
      for (int off = 16; off > 0; off >>= 1) {
        const int sl = (lane ^ off) << 2;
        const float ov = __int_as_float(__builtin_amdgcn_ds_bpermute(sl, __float_as_int(v2)));
        const int   oi = __builtin_amdgcn_ds_bpermute(sl, i2);
        amax2(v2, i2, ov, oi);
      }
      if (lane == 0) { win[0] = v2; win[1] = __int_as_float(i2); }
    }
    __syncthreads();
    const float wv = win[0];
    const int   wi = __float_as_int(win[1]);

    // owner publishes its raw box + area and kills its own score (.at[i].set)
    if (tid == (wi & (TPB - 1))) {
      const int kk = wi >> 10;
#pragma unroll
      for (int k = 0; k < ITEMS; ++k)
        if (k == kk) {
          win[2] = by0[k]; win[3] = bx0[k]; win[4] = by1[k]; win[5] = bx1[k];
          win[6] = ar[k];
          sc[k] = NEGV;
        }
    }
    __syncthreads();
    const float wy0 = win[2], wx0 = win[3], wy1 = win[4], wx1 = win[5], wa = win[6];
    const float Wy1 = fminf(wy0, wy1), Wy2 = fmaxf(wy0, wy1);
    const float Wx1 = fminf(wx0, wx1), Wx2 = fmaxf(wx0, wx1);
    const bool valid = wv > 0.5f;

    if (tid == 0) {
      float* o = out + ((size_t)b * MAXB + it) * 6;
      o[0] = valid ? fminf(fmaxf(wy0, 0.f), 1.f) : 0.f;   // clip raw box
      o[1] = valid ? fminf(fmaxf(wx0, 0.f), 1.f) : 0.f;
      o[2] = valid ? fminf(fmaxf(wy1, 0.f), 1.f) : 0.f;
      o[3] = valid ? fminf(fmaxf(wx1, 0.f), 1.f) : 0.f;
      o[4] = valid ? wv : 0.f;
      o[5] = 0.f;                                          // class id 0
      count += valid ? 1 : 0;
    }

    // suppress IoU > 0.5 against winner (register-only)
#pragma unroll
    for (int k = 0; k < ITEMS; ++k) {
      const float y1 = fminf(by0[k], by1[k]), y2 = fmaxf(by0[k], by1[k]);
      const float x1 = fminf(bx0[k], bx1[k]), x2 = fmaxf(bx0[k], bx1[k]);
      const float iy1 = fmaxf(y1, Wy1), iy2 = fminf(y2, Wy2);
      const float ix1 = fmaxf(x1, Wx1), ix2 = fminf(x2, Wx2);
      const float inter = fmaxf(iy2 - iy1, 0.f) * fmaxf(ix2 - ix1, 0.f);
      const float uni = ar[k] + wa - inter;
      const float iou = (uni > 0.f) ? inter / uni : 0.f;
      if (iou > 0.5f) sc[k] = NEGV;
    }
  }

  if (tid == 0) out[(size_t)gridDim.x * MAXB * 6 + b] = (float)count;
}

extern "C" void kernel_launch(void* const* d_in, const int* in_sizes, int n_in,
                              void* d_out, int out_size, void* d_ws, size_t ws_size,
                              hipStream_t stream) {
  (void)n_in; (void)out_size; (void)d_ws; (void)ws_size;
  const float* bbox13 = (const float*)d_in[0];
  const float* p13    = (const float*)d_in[1];
  const float* c13    = (const float*)d_in[2];
  const float* bbox26 = (const float*)d_in[3];
  const float* p26    = (const float*)d_in[4];
  const float* c26    = (const float*)d_in[5];
  const float* bbox52 = (const float*)d_in[6];
  const float* p52    = (const float*)d_in[7];
  const float* c52    = (const float*)d_in[8];
  const int B = in_sizes[0] / (13 * 13 * 3 * 4);   // 64
  yolo_nms_kernel<<<B, TPB, 0, stream>>>(bbox13, p13, c13,
                                         bbox26, p26, c26,
                                         bbox52, p52, c52,
                                         (float*)d_out);
}